// QGravNet_11819749998725
// MI455X (gfx1250) — compile-verified
//
#include <hip/hip_runtime.h>
#include <hip/hip_bf16.h>

#define BATCH 4
#define NPTS 4096
#define FIN 64
#define NDIM 4
#define NPROP 64
#define NFILT 128
#define KSEL 40          // top-k selected (first is self, dropped)
#define KNBR 39          // actual neighbors kept
#define EXPF 10.0f

typedef __attribute__((ext_vector_type(2))) float v2f;
typedef __attribute__((ext_vector_type(8))) float v8f;

// ---------------------------------------------------------------------------
// fp32 WMMA 16x16x4 helpers (CDNA5 V_WMMA_F32_16X16X4_F32)
//
// A fragment (16x4, row-major source A[M][K], lda):
//   lanes 0-15 : a.x = A[m][kb+0], a.y = A[m][kb+1]   (m = lane)
//   lanes 16-31: a.x = A[m][kb+2], a.y = A[m][kb+3]   (m = lane-16)
// B fragment (4x16, row-major source B[K][N], ldb):
//   lanes 0-15 : b.x = B[kb+0][n], b.y = B[kb+1][n]   (n = lane)
//   lanes 16-31: b.x = B[kb+2][n], b.y = B[kb+3][n]   (n = lane-16)
// C/D 16x16: acc[i] = C[ i + (lane<16 ? 0 : 8) ][ lane & 15 ]
// ---------------------------------------------------------------------------
__device__ __forceinline__ v8f wmma_f32_4(v2f a, v2f b, v8f c) {
    return __builtin_amdgcn_wmma_f32_16x16x4_f32(
        /*neg_a=*/false, a, /*neg_b=*/false, b,
        /*c_mod=*/(short)0, c, /*reuse_a=*/false, /*reuse_b=*/false);
}

__device__ __forceinline__ v2f load_a_frag(const float* A, int lda, int row0,
                                           int kb, int lane) {
    int m  = lane & 15;
    int ko = (lane < 16) ? 0 : 2;
    const float* p = A + (size_t)(row0 + m) * lda + kb + ko;
    v2f a; a.x = p[0]; a.y = p[1];
    return a;
}

__device__ __forceinline__ v2f load_b_frag(const float* Bm, int ldb, int col0,
                                           int kb, int lane) {
    int n  = lane & 15;
    int kk = (lane < 16) ? kb : kb + 2;
    v2f b;
    b.x = Bm[(size_t)kk       * ldb + col0 + n];
    b.y = Bm[(size_t)(kk + 1) * ldb + col0 + n];
    return b;
}

// ---------------------------------------------------------------------------
// Kernel 1: coordinates = x @ W_s + b_s   (16384 x 4, K=64) — plain FMA
// ---------------------------------------------------------------------------
__global__ void qgn_coords(const float* __restrict__ x,
                           const float* __restrict__ Ws,
                           const float* __restrict__ bs,
                           float* __restrict__ coords) {
    int row = blockIdx.x * blockDim.x + threadIdx.x;
    if (row >= BATCH * NPTS) return;
    const float* xr = x + (size_t)row * FIN;
    float c0 = bs[0], c1 = bs[1], c2 = bs[2], c3 = bs[3];
    for (int k = 0; k < FIN; ++k) {
        float xv = xr[k];
        const float* w = Ws + (size_t)k * NDIM;
        c0 = fmaf(xv, w[0], c0);
        c1 = fmaf(xv, w[1], c1);
        c2 = fmaf(xv, w[2], c2);
        c3 = fmaf(xv, w[3], c3);
    }
    float* cr = coords + (size_t)row * NDIM;
    cr[0] = c0; cr[1] = c1; cr[2] = c2; cr[3] = c3;
}

// ---------------------------------------------------------------------------
// Kernel 2: features = x @ W_flr + b_flr  (16384 x 64, K=64) — fp32 WMMA
// block = 256 (8 waves); each wave: one 16-row tile x four 16-col tiles
// ---------------------------------------------------------------------------
__global__ void qgn_feat_gemm(const float* __restrict__ x,
                              const float* __restrict__ Wf,
                              const float* __restrict__ bf,
                              float* __restrict__ feat) {
    int lane = threadIdx.x & 31;
    int wave = threadIdx.x >> 5;
    int row0 = (blockIdx.x * 8 + wave) * 16;

    int mbase = (lane < 16) ? 0 : 8;
    int nloc  = lane & 15;

    for (int ct = 0; ct < NPROP / 16; ++ct) {
        int col0 = ct * 16;
        v8f acc = {};
        #pragma unroll
        for (int kb = 0; kb < FIN; kb += 4) {
            v2f a = load_a_frag(x,  FIN,   row0, kb, lane);
            v2f b = load_b_frag(Wf, NPROP, col0, kb, lane);
            acc = wmma_f32_4(a, b, acc);
        }
        float bias = bf[col0 + nloc];
        #pragma unroll
        for (int i = 0; i < 8; ++i) {
            int row = row0 + mbase + i;
            feat[(size_t)row * NPROP + col0 + nloc] = acc[i] + bias;
        }
    }
}

// ---------------------------------------------------------------------------
// Kernel 3: per-node kNN in 4-D coordinate space.
// One 256-thread block per node. Distances into LDS (16 KB), then 40
// iterative argmin extractions (lowest-index tie-break, matching jax top_k).
// First extraction (self) is dropped; store 39 neighbor idx + exp weights.
// ---------------------------------------------------------------------------
__global__ void qgn_topk(const float* __restrict__ coords,
                         int*   __restrict__ nbr_idx,
                         float* __restrict__ nbr_w) {
    __shared__ float dist[NPTS];
    __shared__ float sval[256];
    __shared__ int   sidx[256];

    const int node = blockIdx.x;          // flattened b*N + n
    const int b    = node / NPTS;
    const int tid  = threadIdx.x;

    const float* cb = coords + (size_t)b * NPTS * NDIM;
    const float* cn = coords + (size_t)node * NDIM;
    const float a0 = cn[0], a1 = cn[1], a2 = cn[2], a3 = cn[3];
    const float na = a0 * a0 + a1 * a1 + a2 * a2 + a3 * a3;

    // Faithful |-2 a.b + |a|^2 + |b|^2| (matches reference fp ordering)
    for (int m = tid; m < NPTS; m += 256) {
        const float* cm = cb + (size_t)m * NDIM;
        float m0 = cm[0], m1 = cm[1], m2 = cm[2], m3 = cm[3];
        float dot = a0 * m0 + a1 * m1 + a2 * m2 + a3 * m3;
        float nb  = m0 * m0 + m1 * m1 + m2 * m2 + m3 * m3;
        dist[m] = fabsf(-2.0f * dot + na + nb);
    }
    __syncthreads();

    for (int it = 0; it < KSEL; ++it) {
        float best = __builtin_inff();
        int   bidx = 0x7fffffff;
        for (int m = tid; m < NPTS; m += 256) {
            float d = dist[m];
            if (d < best) { best = d; bidx = m; }   // ascending m => lowest idx on ties
        }
        sval[tid] = best; sidx[tid] = bidx;
        __syncthreads();
        for (int s = 128; s > 0; s >>= 1) {
            if (tid < s) {
                float v2 = sval[tid + s]; int i2 = sidx[tid + s];
                if (v2 < sval[tid] || (v2 == sval[tid] && i2 < sidx[tid])) {
                    sval[tid] = v2; sidx[tid] = i2;
                }
            }
            __syncthreads();
        }
        if (tid == 0) {
            int   w = sidx[0];
            float d = sval[0];
            if (it > 0) {
                nbr_idx[(size_t)node * KNBR + (it - 1)] = w;
                nbr_w  [(size_t)node * KNBR + (it - 1)] = __expf(-fabsf(d * EXPF));
            }
            dist[w] = __builtin_inff();
        }
        __syncthreads();
    }
}

// ---------------------------------------------------------------------------
// Kernel 4: gather + weighted max/mean aggregation, build updated (.,192)
// 64 threads per node; thread f owns feature channel f.
// ---------------------------------------------------------------------------
__global__ void qgn_agg(const float* __restrict__ x,
                        const float* __restrict__ feat,
                        const int*   __restrict__ nbr_idx,
                        const float* __restrict__ nbr_w,
                        float* __restrict__ updated) {
    const int node = blockIdx.x;
    const int b    = node / NPTS;
    const int f    = threadIdx.x;

    const float* fb = feat + (size_t)b * NPTS * NPROP;
    const int*   ni = nbr_idx + (size_t)node * KNBR;
    const float* nw = nbr_w   + (size_t)node * KNBR;

    float mx = -__builtin_inff();
    float sm = 0.0f;
    #pragma unroll 1
    for (int j = 0; j < KNBR; ++j) {
        float v = fb[(size_t)ni[j] * NPROP + f] * nw[j];
        mx = fmaxf(mx, v);
        sm += v;
    }
    float* up = updated + (size_t)node * (FIN + 2 * NPROP);
    up[f]               = x[(size_t)node * FIN + f];
    up[NPROP + f]       = mx;
    up[2 * NPROP + f]   = sm * (1.0f / (float)KNBR);
}

// ---------------------------------------------------------------------------
// Kernel 5: out = updated @ W_out + b_out  (16384 x 128, K=192) — fp32 WMMA
// block = 256 (8 waves); wave = one 16x16 output tile, 48 WMMA K-steps.
// ---------------------------------------------------------------------------
__global__ void qgn_out_gemm(const float* __restrict__ updated,
                             const float* __restrict__ Wout,
                             const float* __restrict__ bout,
                             float* __restrict__ out) {
    const int KTOT = FIN + 2 * NPROP;     // 192
    int lane = threadIdx.x & 31;
    int wave = threadIdx.x >> 5;
    int gw   = blockIdx.x * 8 + wave;     // 8192 tiles = 1024 row-tiles x 8 col-tiles
    int row0 = (gw >> 3) * 16;
    int col0 = (gw & 7) * 16;

    v8f acc = {};
    #pragma unroll 4
    for (int kb = 0; kb < KTOT; kb += 4) {
        v2f a = load_a_frag(updated, KTOT,  row0, kb, lane);
        v2f b = load_b_frag(Wout,    NFILT, col0, kb, lane);
        acc = wmma_f32_4(a, b, acc);
    }

    int   mbase = (lane < 16) ? 0 : 8;
    int   nloc  = lane & 15;
    float bias  = bout[col0 + nloc];
    #pragma unroll
    for (int i = 0; i < 8; ++i) {
        int row = row0 + mbase + i;
        out[(size_t)row * NFILT + col0 + nloc] = acc[i] + bias;
    }
}

// ---------------------------------------------------------------------------
// Host launcher
// ---------------------------------------------------------------------------
extern "C" void kernel_launch(void* const* d_in, const int* in_sizes, int n_in,
                              void* d_out, int out_size, void* d_ws, size_t ws_size,
                              hipStream_t stream) {
    const float* x     = (const float*)d_in[0];
    const float* W_flr = (const float*)d_in[1];
    const float* b_flr = (const float*)d_in[2];
    const float* W_s   = (const float*)d_in[3];
    const float* b_s   = (const float*)d_in[4];
    const float* W_out = (const float*)d_in[5];
    const float* b_out = (const float*)d_in[6];
    float* out = (float*)d_out;

    const size_t ROWS = (size_t)BATCH * NPTS;       // 16384
    char* ws = (char*)d_ws;
    float* feat    = (float*)(ws);                                   // 16384*64  f32
    float* coords  = (float*)(ws + ROWS * FIN * 4);                  // 16384*4   f32
    int*   nbr_idx = (int*)  (ws + ROWS * (FIN + NDIM) * 4);         // 16384*39  i32
    float* nbr_w   = (float*)(ws + ROWS * (FIN + NDIM + KNBR) * 4);  // 16384*39  f32
    float* updated = (float*)(ws + ROWS * (FIN + NDIM + 2*KNBR) * 4);// 16384*192 f32

    // 1. coordinates (16384 threads)
    qgn_coords<<<(ROWS + 255) / 256, 256, 0, stream>>>(x, W_s, b_s, coords);

    // 2. features GEMM: 1024 row-tiles / 8 waves per block = 128 blocks
    qgn_feat_gemm<<<(ROWS / 16) / 8, 256, 0, stream>>>(x, W_flr, b_flr, feat);

    // 3. kNN top-40 per node
    qgn_topk<<<ROWS, 256, 0, stream>>>(coords, nbr_idx, nbr_w);

    // 4. aggregation -> updated
    qgn_agg<<<ROWS, FIN, 0, stream>>>(x, feat, nbr_idx, nbr_w, updated);

    // 5. output GEMM: 1024 row-tiles * 8 col-tiles / 8 waves per block
    qgn_out_gemm<<<(ROWS / 16), 256, 0, stream>>>(updated, W_out, b_out, out);
}